// TaylorBaseEBM_54717883351298
// MI455X (gfx1250) — compile-verified
//
#include <hip/hip_runtime.h>

typedef __attribute__((ext_vector_type(2))) float v2f;
typedef __attribute__((ext_vector_type(8))) float v8f;

#define STEPS 20

__device__ __forceinline__ float fast_rcp(float x) {
  return __builtin_amdgcn_rcpf(x);
}

__device__ __forceinline__ float swap16(float x) {
  int i = __builtin_bit_cast(int, x);
  int r = __builtin_amdgcn_permlanex16(i, i, 0x76543210, 0xfedcba98, false, false);
  return __builtin_bit_cast(float, r);
}

// Column permutation that lets layer-l D registers be reused directly as
// layer-(l+1) B registers:  colp(k) = mD(kB^-1(k)).
//   kB(reg j, half h) = 4*(j/2) + (j&1) + 2h   (assumed HW B layout, 4x16 f32)
//   mD(reg j, half h) = 16*(j/8) + (j&7) + 8h  (HW D layout, 16x16 f32)
__device__ __forceinline__ int colp(int k) {
  int j = 2 * (k >> 2) + (k & 1);
  int h = (k >> 1) & 1;
  return ((j >> 3) << 4) + (j & 7) + (h << 3);
}

// swish jet coefficients:
//   v0 = z*sig, s' = sig*(1+u) with u = z*(1-sig),
//   s'' = sig*(1-sig)*(2 + z*(1-2sig)) = (sig*t)*(2 + u - v0), t = e*sig = 1-sig
__device__ __forceinline__ void swish_coeffs(float z0, float& v0, float& sp, float& spp) {
  float e   = __expf(-z0);
  float sig = fast_rcp(1.0f + e);       // v_rcp_f32
  float t   = e * sig;                  // 1 - sig
  v0  = z0 * sig;
  float u   = z0 * t;
  sp  = fmaf(sig, u, sig);
  spp = (sig * t) * ((u - v0) + 2.0f);
}

// One hidden layer: 3 jet channels, z_c = W * h_c (+ bias on channel 0).
// in* are B-operand pairs (K rows, column-permuted order), out* receive the
// activated jets at D positions (which are the next layer's B positions).
// NEED_V=false skips the dead value-channel output (last hidden layer).
template <bool NEED_V>
__device__ __forceinline__ void hidden_layer(
    const v2f (&in0)[12], const v2f (&in1)[12], const v2f (&in2)[12],
    v2f (&o0)[12], v2f (&o1)[12], v2f (&o2)[12],
    const float (*lwl)[12][32][2],      // lw[l]: [mt][kg][lane][2]
    const float (*cb)[8][2],            // bias, D layout [mt][r][half]
    int lane, int hi) {
#pragma unroll
  for (int mt = 0; mt < 3; ++mt) {
    v8f a0, a1, a2;
#pragma unroll
    for (int i = 0; i < 8; ++i) {
      a0[i] = cb[mt][i][hi];            // bias rides in the C operand
      a1[i] = 0.f;
      a2[i] = 0.f;
    }
#pragma unroll
    for (int kg = 0; kg < 12; ++kg) {
      v2f A = *(const v2f*)(&lwl[mt][kg][lane][0]);   // ds_load_b64, 3 WMMAs
      a0 = __builtin_amdgcn_wmma_f32_16x16x4_f32(false, A, false, in0[kg], (short)0, a0, false, false);
      a1 = __builtin_amdgcn_wmma_f32_16x16x4_f32(false, A, false, in1[kg], (short)0, a1, false, false);
      a2 = __builtin_amdgcn_wmma_f32_16x16x4_f32(false, A, false, in2[kg], (short)0, a2, false, false);
    }
#pragma unroll
    for (int i = 0; i < 8; ++i) {
      const int j = mt * 8 + i;         // D reg index == next-layer B reg index
      float v0, sp, spp;
      swish_coeffs(a0[i], v0, sp, spp);
      float z1 = a1[i], z2 = a2[i];
      if (NEED_V) o0[j >> 1][j & 1] = v0;
      o1[j >> 1][j & 1] = sp * z1;
      o2[j >> 1][j & 1] = fmaf(spp * z1, z1, sp * z2);
    }
  }
}

__global__ __launch_bounds__(256) void taylor_ebm_kernel(
    const float* __restrict__ x_g, const float* __restrict__ W0,
    const float* __restrict__ b0g, const float* __restrict__ W1,
    const float* __restrict__ b1g, const float* __restrict__ W2,
    const float* __restrict__ b2g, const float* __restrict__ W3,
    const float* __restrict__ eps_g, float* __restrict__ out, int nB) {
  // A-fragments for W1/W2 with permuted columns:
  // lw[l][mt][kg][lane] = { W[row][colp(kb)], W[row][colp(kb+1)] },
  // row = mt*16 + (lane&15), kb = 4*kg + 2*(lane>>4)  (ISA 16x4 f32 A layout)
  __shared__ float lw[2][3][12][32][2];
  __shared__ float cw01[3][8][2];   // W0[:,1] in D layout [mt][r][half]
  __shared__ float cb1[3][8][2];    // b1 in D layout
  __shared__ float cb2[3][8][2];    // b2 in D layout
  __shared__ float cw3[3][8][2];    // W3 in D layout

  const int tid = threadIdx.x;
  for (int idx = tid; idx < 2 * 3 * 12 * 32; idx += 256) {
    int l   = idx / (3 * 12 * 32);
    int rem = idx - l * (3 * 12 * 32);
    int mt  = rem / (12 * 32);
    int r2  = rem - mt * (12 * 32);
    int kg  = r2 >> 5;
    int ln  = r2 & 31;
    int row = mt * 16 + (ln & 15);
    int kb  = 4 * kg + ((ln >> 4) << 1);
    const float* W = l ? W2 : W1;
    lw[l][mt][kg][ln][0] = W[row * 48 + colp(kb)];
    lw[l][mt][kg][ln][1] = W[row * 48 + colp(kb + 1)];
  }
  if (tid < 48) {                       // D-layout constants (unit = mt*16+r+8h)
    int mt = tid >> 4, rr = tid & 15, r = rr & 7, h = rr >> 3;
    int u  = mt * 16 + r + h * 8;
    cw01[mt][r][h] = W0[2 * u + 1];
    cb1[mt][r][h]  = b1g[u];
    cb2[mt][r][h]  = b2g[u];
    cw3[mt][r][h]  = W3[u];
  }
  __syncthreads();

  const int lane = tid & 31;
  const int wave = tid >> 5;
  const int hi   = lane >> 4;                                    // lane-half
  const int elem = (blockIdx.x * 8 + wave) * 16 + (lane & 15);   // 16 elems/wave

  const float x = x_g[elem];

  // layer-0 step-invariant jet base: W0[:,0]*x + b0, in D layout
  float caF[3][8];
#pragma unroll
  for (int mt = 0; mt < 3; ++mt)
#pragma unroll
    for (int r = 0; r < 8; ++r) {
      int u = mt * 16 + r + hi * 8;
      caF[mt][r] = W0[2 * u] * x + b0g[u];
    }

  float y = 0.f, am = 0.f, av = 0.f;
  float bp1 = 1.f, bp2 = 1.f;

#pragma unroll 1
  for (int s = 0; s < STEPS; ++s) {
    float eps = eps_g[s * nB + elem];
    float y0  = y + eps;

    // double-buffered jet state, stored directly as B-operand pairs
    v2f ha0[12], ha1[12], ha2[12];
    v2f hb0[12], hb1[12], hb2[12];

    // layer 0: K=2, input jet ([x,y0],[0,1],[0,0]) -> z1 = W0[:,1], z2 = 0
#pragma unroll
    for (int mt = 0; mt < 3; ++mt)
#pragma unroll
      for (int r = 0; r < 8; ++r) {
        const int j = mt * 8 + r;
        float w01 = cw01[mt][r][hi];
        float z0  = caF[mt][r] + w01 * y0;
        float v0, sp, spp;
        swish_coeffs(z0, v0, sp, spp);
        ha0[j >> 1][j & 1] = v0;
        ha1[j >> 1][j & 1] = sp * w01;
        ha2[j >> 1][j & 1] = (spp * w01) * w01;   // z2 = 0 path
      }

    // hidden layers (column-permuted weights -> no relayout between layers)
    hidden_layer<true >(ha0, ha1, ha2, hb0, hb1, hb2, lw[0], cb1, lane, hi);
    hidden_layer<false>(hb0, hb1, hb2, ha0, ha1, ha2, lw[1], cb2, lane, hi);

    // output layer in D layout: only g1 = W3.h1, g2 = W3.h2 matter
    float p1 = 0.f, p2 = 0.f;
#pragma unroll
    for (int mt = 0; mt < 3; ++mt)
#pragma unroll
      for (int r = 0; r < 8; ++r) {
        const int j = mt * 8 + r;
        float w3 = cw3[mt][r][hi];
        p1 = fmaf(w3, ha1[j >> 1][j & 1], p1);
        p2 = fmaf(w3, ha2[j >> 1][j & 1], p2);
      }
    float g1 = p1 + swap16(p1);   // halves hold disjoint unit subsets
    float g2 = p2 + swap16(p2);
    float g  = g1 - g2 * eps;     // f'(y0) + f''(y0)*(y - y0), y - y0 = -eps

    // Adam (duplicated across lane halves, consistent)
    bp1 *= 0.9f;  bp2 *= 0.999f;
    am = 0.9f * am + 0.1f * g;
    av = 0.999f * av + 0.001f * g * g;
    float mh = am * fast_rcp(1.0f - bp1);
    float vh = av * fast_rcp(1.0f - bp2);
    y = y - 0.1f * mh * fast_rcp(__builtin_amdgcn_sqrtf(vh) + 1e-8f);
  }

  if (lane < 16) out[elem] = y;
}

extern "C" void kernel_launch(void* const* d_in, const int* in_sizes, int n_in,
                              void* d_out, int out_size, void* d_ws, size_t ws_size,
                              hipStream_t stream) {
  (void)n_in; (void)out_size; (void)d_ws; (void)ws_size;
  const float* x   = (const float*)d_in[0];
  const float* W0  = (const float*)d_in[1];
  const float* b0  = (const float*)d_in[2];
  const float* W1  = (const float*)d_in[3];
  const float* b1  = (const float*)d_in[4];
  const float* W2  = (const float*)d_in[5];
  const float* b2  = (const float*)d_in[6];
  const float* W3  = (const float*)d_in[7];
  // d_in[8] = b3: constant offset, drops out of the gradient
  const float* eps = (const float*)d_in[9];
  float* out = (float*)d_out;

  int nB = in_sizes[0];                 // 131072
  int blocks = nB / 128;                // 8 waves/block, 16 elems/wave
  taylor_ebm_kernel<<<blocks, 256, 0, stream>>>(x, W0, b0, W1, b1, W2, b2, W3,
                                                eps, out, nB);
}